// GCN_47691316854949
// MI455X (gfx1250) — compile-verified
//
#include <hip/hip_runtime.h>

#define N_NODES 50000
#define N_EDGES 800000
#define IN_CH   512
#define HID     256

#define BM 128
#define BN 32
#define KT 32
#define LDA_S 36   // padded LDS row stride (floats): 16B-aligned rows, conflict-spread banks
#define BFRAG  1024 // B tile in fragment order: [8 steps][2 pairs][32 lanes][2]

typedef float v2f __attribute__((ext_vector_type(2)));
typedef float v8f __attribute__((ext_vector_type(8)));
typedef int   v4i __attribute__((ext_vector_type(4)));

#if defined(__has_builtin)
#if __has_builtin(__builtin_amdgcn_global_load_async_to_lds_b128)
#define HAVE_ASYNC_LDS 1
#endif
#if __has_builtin(__builtin_amdgcn_s_wait_asynccnt)
#define HAVE_WAIT_ASYNC 1
#endif
#endif

#if defined(HAVE_ASYNC_LDS)
typedef __attribute__((address_space(1))) v4i gv4i;   // global int4
typedef __attribute__((address_space(3))) v4i lv4i;   // LDS int4
#endif

static __device__ __forceinline__ void wait_async0() {
#if defined(HAVE_ASYNC_LDS) && defined(HAVE_WAIT_ASYNC)
    __builtin_amdgcn_s_wait_asynccnt(0);
#endif
}

// ---------------- degree / norm ----------------

__global__ void k_deg(const int* __restrict__ dst, float* __restrict__ deg, int E) {
    int t = blockIdx.x * blockDim.x + threadIdx.x;
    if (t < E) atomicAdd(&deg[dst[t]], 1.0f);
}

__global__ void k_dinv(float* __restrict__ deg, int n) {
    int t = blockIdx.x * blockDim.x + threadIdx.x;
    if (t < n) deg[t] = rsqrtf(deg[t] + 1.0f);   // +1 self loop
}

// ---------------- fp32 WMMA GEMM, double-buffered LDS, async A staging ----------------
// C[M,Nstore] = A[M,Kp] @ B[Kreal,Nreal] (+bias), 16x16x4 f32 WMMA.
// 256-thread block -> 128x32 tile; wave w -> rows [w*16,w*16+16), two 16x16 accs.
// A: async DMA to LDS (rows clamped; K-padding guaranteed by zero-padded buffers).
// B: register-staged into LDS in *fragment order* so each lane frag is one b64 load.
__global__ __launch_bounds__(256) void k_wmma_gemm(
        const float* __restrict__ A, int lda,
        const float* __restrict__ B, int ldb,
        const float* __restrict__ bias,
        float* __restrict__ C, int ldc,
        int M, int Kp, int Kreal, int Nreal, int Nstore) {
    __shared__ float As[2][BM * LDA_S];
    __shared__ float Bs[2][BFRAG];

    const int tid  = threadIdx.x;
    const int wave = tid >> 5;
    const int lane = tid & 31;
    const int half = lane >> 4;
    const int l16  = lane & 15;
    const int m0   = blockIdx.x * BM;
    const int n0   = blockIdx.y * BN;
    const int nTiles = Kp / KT;                   // Kp is always a multiple of 32

    // A staging: 2 threads per row, 16 consecutive k (4x b128) each.
    const int ar = tid >> 1;
    const int ak = (tid & 1) * 16;
    int arow_g = m0 + ar;
    if (arow_g >= M) arow_g = M - 1;              // clamp: OOB rows discarded at store
    const float* Ag = A + (size_t)arow_g * lda + ak;

    // B staging coords: k = tid>>3 (0..31), cols (tid&7)*4..+3, zero-filled.
    const int bk  = tid >> 3;
    const int bq  = bk & 3;                       // k within group of 4
    const int bsS = bk >> 2;                      // step index
    const int bc0 = (tid & 7) * 4;

    auto stageA = [&](int k0, int buf) {
        float* dstp = &As[buf][ar * LDA_S + ak];
#if defined(HAVE_ASYNC_LDS)
#pragma unroll
        for (int j = 0; j < 4; ++j)
            __builtin_amdgcn_global_load_async_to_lds_b128(
                (gv4i*)(Ag + k0 + 4 * j), (lv4i*)(dstp + 4 * j), 0, 0);
#else
#pragma unroll
        for (int j = 0; j < 4; ++j)
            *(float4*)(dstp + 4 * j) = *(const float4*)(Ag + k0 + 4 * j);
#endif
    };
    // fragment-order scatter: value (k=4s+q, c) -> ((s*2 + c/16)*32 + (q&1)*16 + (c&15))*2 + (q>>1)
    auto stageB = [&](int k0, int buf) {
#pragma unroll
        for (int j = 0; j < 4; ++j) {
            const int kk = k0 + bk;
            const int cc = n0 + bc0 + j;
            const float v = (kk < Kreal && cc < Nreal) ? B[(size_t)kk * ldb + cc] : 0.f;
            const int c = bc0 + j;
            const int off = ((bsS * 2 + (c >> 4)) * 32 + (bq & 1) * 16 + (c & 15)) * 2 + (bq >> 1);
            Bs[buf][off] = v;
        }
    };

    v8f acc0 = {0.f,0.f,0.f,0.f,0.f,0.f,0.f,0.f};
    v8f acc1 = {0.f,0.f,0.f,0.f,0.f,0.f,0.f,0.f};

    // prologue: tile 0 into buffer 0
    stageA(0, 0);
    stageB(0, 0);
    wait_async0();
    __syncthreads();

    for (int t = 0; t < nTiles; ++t) {
        const int cur = t & 1;
        if (t + 1 < nTiles) {                     // prefetch next tile into other buffer
            stageA((t + 1) * KT, cur ^ 1);
            stageB((t + 1) * KT, cur ^ 1);
        }

#pragma unroll
        for (int s = 0; s < 8; ++s) {
            const v2f a  = *(const v2f*)&As[cur][(wave * 16 + l16) * LDA_S + 4 * s + 2 * half];
            const v2f b0 = *(const v2f*)&Bs[cur][((s * 2 + 0) * 32 + lane) * 2];
            const v2f b1 = *(const v2f*)&Bs[cur][((s * 2 + 1) * 32 + lane) * 2];
            acc0 = __builtin_amdgcn_wmma_f32_16x16x4_f32(false, a, false, b0,
                                                         (short)0, acc0, false, false);
            acc1 = __builtin_amdgcn_wmma_f32_16x16x4_f32(false, a, false, b1,
                                                         (short)0, acc1, false, false);
        }
        wait_async0();          // next tile's async A-writes landed
        __syncthreads();        // all waves done reading `cur` (dscnt flushed pre-signal)
    }

    // ---- store (C/D layout: VGPR r -> M = r + 8*half, N = l16) ----
    const int colA = n0 + l16;
    const int colB = n0 + 16 + l16;
    const float bvA = (bias && colA < Nreal) ? bias[colA] : 0.f;
    const float bvB = (bias && colB < Nreal) ? bias[colB] : 0.f;
#pragma unroll
    for (int r = 0; r < 8; ++r) {
        const int row = m0 + wave * 16 + half * 8 + r;
        if (row < M) {
            if (colA < Nstore) C[(size_t)row * ldc + colA] = acc0[r] + bvA;
            if (colB < Nstore) C[(size_t)row * ldc + colB] = acc1[r] + bvB;
        }
    }
}

// ------- edge scatter-add (float4 gather): agg[dst] += xw[src] * dinv[src]*dinv[dst] -------

__global__ void k_edge_agg4(const int* __restrict__ src, const int* __restrict__ dst,
                            const float* __restrict__ xw, const float* __restrict__ dinv,
                            float* __restrict__ agg, int E, int C4, int C) {
    long t = (long)blockIdx.x * blockDim.x + threadIdx.x;
    if (t >= (long)E * C4) return;
    const int e = (int)(t / C4);
    const int q = (int)(t - (long)e * C4) * 4;
    const int s = src[e], d = dst[e];
    const float nrm = dinv[s] * dinv[d];
    const float4 v = *(const float4*)&xw[(size_t)s * C + q];
    float* o = &agg[(size_t)d * C + q];
    atomicAdd(o + 0, v.x * nrm);
    atomicAdd(o + 1, v.y * nrm);
    atomicAdd(o + 2, v.z * nrm);
    atomicAdd(o + 3, v.w * nrm);
}

// ---------------- finalize: agg <- act(agg + xw*dinv^2 + b) (in place, strided) ----------------

__global__ void k_finalize(float* __restrict__ agg, const float* __restrict__ xw,
                           const float* __restrict__ dinv, const float* __restrict__ b,
                           int n, int C, int ldx, int do_relu) {
    long t = (long)blockIdx.x * blockDim.x + threadIdx.x;
    if (t >= (long)n * C) return;
    const int i = (int)(t / C);
    const int c = (int)(t - (long)i * C);
    const size_t idx = (size_t)i * ldx + c;
    const float di = dinv[i];
    const float v = agg[idx] + xw[idx] * di * di + b[c];
    agg[idx] = do_relu ? fmaxf(v, 0.f) : v;
}

// ---------------- row softmax, in place ----------------

__global__ void k_softmax(float* __restrict__ out, int n, int C) {
    int i = blockIdx.x * blockDim.x + threadIdx.x;
    if (i >= n) return;
    float* row = out + (size_t)i * C;
    float m = -3.402823466e38f;
    for (int c = 0; c < C; ++c) m = fmaxf(m, row[c]);
    float s = 0.f;
    for (int c = 0; c < C; ++c) { float e = __expf(row[c] - m); row[c] = e; s += e; }
    const float inv = 1.0f / s;
    for (int c = 0; c < C; ++c) row[c] *= inv;
}

// ---------------- host orchestration ----------------

extern "C" void kernel_launch(void* const* d_in, const int* in_sizes, int n_in,
                              void* d_out, int out_size, void* d_ws, size_t ws_size,
                              hipStream_t stream) {
    const float* x  = (const float*)d_in[0];
    const int*   ei = (const int*)  d_in[1];
    const float* W1 = (const float*)d_in[2];
    const float* b1 = (const float*)d_in[3];
    const int* src = ei;
    const int* dst = ei + N_EDGES;

    // workspace (floats): dinv | xw1 | h | gp(stride 352) | ga(stride 352)
    float* dinv = (float*)d_ws;
    float* xw1  = dinv + N_NODES;
    float* h    = xw1  + (size_t)N_NODES * HID;
    float* gp   = h    + (size_t)N_NODES * HID;
    float* ga   = gp   + (size_t)N_NODES * 352;

    const int mBlocks = (N_NODES + BM - 1) / BM;   // 391

    // degrees -> dinv
    (void)hipMemsetAsync(dinv, 0, N_NODES * sizeof(float), stream);
    k_deg <<<(N_EDGES + 255) / 256, 256, 0, stream>>>(dst, dinv, N_EDGES);
    k_dinv<<<(N_NODES + 255) / 256, 256, 0, stream>>>(dinv, N_NODES);

    // layer 1: xw1 = x @ W1 ; h = relu(agg + xw1*dinv^2 + b1)
    {
        dim3 grid(mBlocks, HID / BN);
        k_wmma_gemm<<<grid, 256, 0, stream>>>(x, IN_CH, W1, HID, nullptr,
                                              xw1, HID, N_NODES, IN_CH, IN_CH, HID, HID);
    }
    (void)hipMemsetAsync(h, 0, (size_t)N_NODES * HID * sizeof(float), stream);
    {
        long total = (long)N_EDGES * (HID / 4);
        k_edge_agg4<<<(unsigned)((total + 255) / 256), 256, 0, stream>>>(
            src, dst, xw1, dinv, h, N_EDGES, HID / 4, HID);
        long tn = (long)N_NODES * HID;
        k_finalize<<<(unsigned)((tn + 255) / 256), 256, 0, stream>>>(
            h, xw1, dinv, b1, N_NODES, HID, HID, 1);
    }

    // four heads
    const int ncs[4]  = {10, 25, 16, 350};
    const int ncps[4] = {32, 32, 32, 352};   // padded strides, multiple of 32
    size_t out_off = 0;
    for (int hd = 0; hd < 4; ++hd) {
        const float* Wg = (const float*)d_in[4 + 4 * hd];
        const float* bg = (const float*)d_in[5 + 4 * hd];
        const float* Wc = (const float*)d_in[6 + 4 * hd];
        const float* bc = (const float*)d_in[7 + 4 * hd];
        const int nc  = ncs[hd];
        const int ncp = ncps[hd];

        // gp[:, :ncp] = h @ Wg  (padding cols get exact zeros: B zero-filled, bias null)
        {
            dim3 grid(mBlocks, ncp / BN);
            k_wmma_gemm<<<grid, 256, 0, stream>>>(h, HID, Wg, nc, nullptr,
                                                  gp, ncp, N_NODES, HID, HID, nc, ncp);
        }
        // ga = segment_sum(gp[src]*norm, dst) over padded stride; finalize real cols
        (void)hipMemsetAsync(ga, 0, (size_t)N_NODES * ncp * sizeof(float), stream);
        {
            long total = (long)N_EDGES * (ncp / 4);
            k_edge_agg4<<<(unsigned)((total + 255) / 256), 256, 0, stream>>>(
                src, dst, gp, dinv, ga, N_EDGES, ncp / 4, ncp);
            long tn = (long)N_NODES * nc;
            k_finalize<<<(unsigned)((tn + 255) / 256), 256, 0, stream>>>(
                ga, gp, dinv, bg, N_NODES, nc, ncp, 0);
        }
        // logits = ga @ Wc + bc -> d_out (dense stride nc); softmax in place
        float* outh = (float*)d_out + out_off;
        {
            dim3 grid(mBlocks, (nc + BN - 1) / BN);
            k_wmma_gemm<<<grid, 256, 0, stream>>>(ga, ncp, Wc, nc, bc,
                                                  outh, nc, N_NODES, ncp, nc, nc, nc);
        }
        k_softmax<<<(N_NODES + 255) / 256, 256, 0, stream>>>(outh, N_NODES, nc);

        out_off += (size_t)N_NODES * nc;
    }
}